// LSTMEncoderLayer_68607807586874
// MI455X (gfx1250) — compile-verified
//
#include <hip/hip_runtime.h>
#include <cstdint>
#include <cstddef>

typedef float v2f __attribute__((ext_vector_type(2)));
typedef float v8f __attribute__((ext_vector_type(8)));

#define NB   64      // batch
#define NT   512     // time steps
#define ND   512     // input dim
#define NH   1024    // hidden dim
#define N4H  4096    // 4*H, gate-concatenated width
#define NK   (ND + NH)  // 1536, fused K

__device__ __forceinline__ float sigmoidf_(float x) {
    return 1.0f / (1.0f + __expf(-x));
}

// ---------------------------------------------------------------------------
// Kernel 1: ZX = x @ W + b      (M = B*T = 32768, K = D = 512, N = 4H = 4096)
// fp32 WMMA 16x16x4. block = 256 threads = 8 waves; waves share the A (x) rows
// and fan out over 8 adjacent 16-wide N tiles.
// grid = (M/16, N/(16*8))
// ---------------------------------------------------------------------------
__global__ __launch_bounds__(256) void zx_gemm_kernel(
    const float* __restrict__ X, const float* __restrict__ W,
    const float* __restrict__ bias, float* __restrict__ ZX)
{
    const int wave = threadIdx.x >> 5;
    const int lane = threadIdx.x & 31;
    const int half = lane >> 4;        // 0: K pair {k,k+1}, 1: {k+2,k+3}
    const int l15  = lane & 15;

    const int mbase = blockIdx.x * 16;
    const int nbase = (blockIdx.y * 8 + wave) * 16;

    const float* aptr = X + (size_t)(mbase + l15) * ND;   // A row for this lane
    const float* bptr = W + nbase + l15;                  // B col (n = l15)

    v8f acc = (v8f)0.0f;
    for (int k = 0; k < ND; k += 4) {
        const int kp = k + 2 * half;
        v2f a, b;
        a.x = aptr[kp];
        a.y = aptr[kp + 1];
        b.x = bptr[(size_t)kp * N4H];
        b.y = bptr[(size_t)(kp + 1) * N4H];
        acc = __builtin_amdgcn_wmma_f32_16x16x4_f32(
            false, a, false, b, (short)0, acc, false, false);
    }

    const float bv = bias[nbase + l15];
    // C/D layout: VGPR r, lane -> m = r + 8*half, n = l15
    float* out = ZX + (size_t)(mbase + 8 * half) * N4H + nbase + l15;
#pragma unroll
    for (int r = 0; r < 8; ++r)
        out[(size_t)r * N4H] = acc[r] + bv;
}

// ---------------------------------------------------------------------------
// Kernel 2: one LSTM time step.
//   z = base + [x_t | h] @ [W ; U]   (K = 1536 fused, or K = 1024 with base=ZX)
// Block owns (16 batch rows) x (16 hidden cols) x all 4 gates, so the gate
// nonlinearity + c/h update is block-local. 8 waves split K, LDS-reduce.
// grid = (B/16=4, H/16=64), block = 256.
// ---------------------------------------------------------------------------
__global__ __launch_bounds__(256) void lstm_step_kernel(
    const float* __restrict__ X,    // x (fused path)
    const float* __restrict__ ZX,   // precomputed x@W+b, or nullptr
    const float* __restrict__ W,    // [D,4H] (fused path)
    const float* __restrict__ U,    // [H,4H]
    const float* __restrict__ bias, // [4H]   (fused path)
    const float* __restrict__ h_in,
    float* __restrict__ h_out,
    float* __restrict__ c_st,
    int t, int k0)                  // k0 = 512 (ZX path) or 0 (fused)
{
    __shared__ float part[8][4][32][8];   // 32 KB: per-wave gate partials
    __shared__ float red[4][32][8];       //  4 KB: reduced gate tiles

    const int wave = threadIdx.x >> 5;
    const int lane = threadIdx.x & 31;
    const int half = lane >> 4;
    const int l15  = lane & 15;

    const int mbase = blockIdx.x * 16;    // batch-row tile
    const int jbase = blockIdx.y * 16;    // hidden-col tile

    const int arow   = mbase + l15;       // batch index for A lane
    const int kslice = (NK - k0) >> 3;    // 128 (ZX) or 192 (fused)
    const int kbeg   = k0 + wave * kslice;
    const int kend   = kbeg + kslice;

    const float* xrow = X + ((size_t)arow * NT + t) * ND;
    const float* hrow = h_in + (size_t)arow * NH;

    v8f acc[4];
#pragma unroll
    for (int g = 0; g < 4; ++g) acc[g] = (v8f)0.0f;

    for (int k = kbeg; k < kend; k += 4) {
        const int kp = k + 2 * half;      // pair never straddles the D boundary
        v2f a;
        if (kp < ND) { a.x = xrow[kp];      a.y = xrow[kp + 1]; }
        else         { a.x = hrow[kp - ND]; a.y = hrow[kp - ND + 1]; }
        const float* brow0 = (kp < ND) ? (W + (size_t)kp * N4H)
                                       : (U + (size_t)(kp - ND) * N4H);
        const float* brow1 = brow0 + N4H; // kp+1 stays on same side of boundary
#pragma unroll
        for (int g = 0; g < 4; ++g) {
            const int col = g * NH + jbase + l15;
            v2f b;
            b.x = brow0[col];
            b.y = brow1[col];
            acc[g] = __builtin_amdgcn_wmma_f32_16x16x4_f32(
                false, a, false, b, (short)0, acc[g], false, false);
        }
    }

#pragma unroll
    for (int g = 0; g < 4; ++g)
#pragma unroll
        for (int r = 0; r < 8; ++r)
            part[wave][g][lane][r] = acc[g][r];
    __syncthreads();

    // waves 0..3 each reduce one gate across the 8 K-slices
    if (wave < 4) {
#pragma unroll
        for (int r = 0; r < 8; ++r) {
            float s = 0.0f;
#pragma unroll
            for (int w = 0; w < 8; ++w) s += part[w][wave][lane][r];
            red[wave][lane][r] = s;
        }
    }
    __syncthreads();

    // pointwise LSTM update: each of the 256 threads owns one (m,j) element
    {
        const int l = threadIdx.x & 31;
        const int r = threadIdx.x >> 5;
        const int m = mbase + r + 8 * (l >> 4);
        const int j = jbase + (l & 15);

        float z[4];
#pragma unroll
        for (int g = 0; g < 4; ++g) {
            float base = ZX ? ZX[((size_t)m * NT + t) * N4H + g * NH + j]
                            : bias[g * NH + j];
            z[g] = red[g][l][r] + base;
        }
        const float ig = sigmoidf_(z[0]);
        const float fg = sigmoidf_(z[1]);
        const float gg = tanhf(z[2]);
        const float og = sigmoidf_(z[3]);

        const size_t idx = (size_t)m * NH + j;
        const float c_new = fg * c_st[idx] + ig * gg;
        c_st[idx]  = c_new;
        h_out[idx] = og * tanhf(c_new);
    }
}

// ---------------------------------------------------------------------------
// Kernel 3: out[B, 3H] = concat(hT, hT, cT)
// ---------------------------------------------------------------------------
__global__ void concat_out_kernel(const float* __restrict__ h,
                                  const float* __restrict__ c,
                                  float* __restrict__ out)
{
    int idx = blockIdx.x * blockDim.x + threadIdx.x;
    if (idx >= NB * 3 * NH) return;
    int row = idx / (3 * NH);
    int col = idx % (3 * NH);
    float v = (col < 2 * NH) ? h[(size_t)row * NH + (col % NH)]
                             : c[(size_t)row * NH + (col - 2 * NH)];
    out[idx] = v;
}

// ---------------------------------------------------------------------------
extern "C" void kernel_launch(void* const* d_in, const int* in_sizes, int n_in,
                              void* d_out, int out_size, void* d_ws, size_t ws_size,
                              hipStream_t stream)
{
    (void)in_sizes; (void)n_in; (void)out_size;
    const float* x    = (const float*)d_in[0];
    const float* W    = (const float*)d_in[1];
    const float* U    = (const float*)d_in[2];
    const float* bias = (const float*)d_in[3];

    float* h0 = (float*)d_ws;
    float* h1 = h0 + NB * NH;
    float* c  = h1 + NB * NH;
    float* zx = c  + NB * NH;

    const size_t state_bytes = (size_t)3 * NB * NH * sizeof(float);
    const size_t zx_bytes    = (size_t)NB * NT * N4H * sizeof(float);
    const bool useZX = ws_size >= state_bytes + zx_bytes;

    hipMemsetAsync(h0, 0, (size_t)NB * NH * sizeof(float), stream);
    hipMemsetAsync(c,  0, (size_t)NB * NH * sizeof(float), stream);

    if (useZX) {
        dim3 grid((NB * NT) / 16, N4H / (16 * 8));
        zx_gemm_kernel<<<grid, 256, 0, stream>>>(x, W, bias, zx);
    }

    for (int t = 0; t < NT; ++t) {
        float* hin  = (t & 1) ? h1 : h0;
        float* hout = (t & 1) ? h0 : h1;
        lstm_step_kernel<<<dim3(NB / 16, NH / 16), 256, 0, stream>>>(
            x, useZX ? zx : nullptr, W, U, bias, hin, hout, c,
            t, useZX ? ND : 0);
    }
    // T=512 is even: last step (t=511) reads h1, writes h0 -> hT lives in h0.
    concat_out_kernel<<<(NB * 3 * NH + 255) / 256, 256, 0, stream>>>(
        h0, c, (float*)d_out);
}